// IntraDecoderAttention_4183298146637
// MI455X (gfx1250) — compile-verified
//
#include <hip/hip_runtime.h>

typedef float v2f __attribute__((ext_vector_type(2)));
typedef float v8f __attribute__((ext_vector_type(8)));

#define B_  32
#define T_  2048
#define H_  1024

// ---------------------------------------------------------------------------
// Kernel A: q[b,h] = sum_o dh[b,o] * W[o,h]   (M=32, N=1024, K=1024)
// One wave per 16x16 output tile, V_WMMA_F32_16X16X4_F32, K-step = 4.
// A-frag 16x4 f32 layout (ISA 7.12.2): lanes 0-15: M=lane, K={0,1} in v0/v1;
// lanes 16-31: M=lane-16, K={2,3}. B-frag 4x16 mirrored over N. C/D: VGPR r
// holds rows r (lanes 0-15) and r+8 (lanes 16-31), N = lane&15.
// ---------------------------------------------------------------------------
__global__ void __launch_bounds__(32)
qgemm_wmma(const float* __restrict__ dh, const float* __restrict__ W,
           float* __restrict__ q) {
  const int tile   = blockIdx.x;          // 128 tiles = 2 (M) x 64 (N)
  const int lane   = threadIdx.x;
  const int mt     = tile & 1;
  const int nt     = tile >> 1;
  const int m_base = mt * 16;
  const int n_base = nt * 16;
  const int half   = lane >> 4;           // 0: K={0,1}, 1: K={2,3}
  const int l15    = lane & 15;
  const int kk     = half * 2;

  const float* arow = dh + (size_t)(m_base + l15) * H_;   // A[m, :]
  const float* bcol = W  + (n_base + l15);                // B[:, n] = W[o, n]

  v8f c = {};
  for (int k0 = 0; k0 < H_; k0 += 4) {
    v2f a, b;
    a.x = arow[k0 + kk];
    a.y = arow[k0 + kk + 1];
    b.x = bcol[(size_t)(k0 + kk)     * H_];
    b.y = bcol[(size_t)(k0 + kk + 1) * H_];
    c = __builtin_amdgcn_wmma_f32_16x16x4_f32(false, a, false, b,
                                              (short)0, c, false, false);
  }
  const int row0 = m_base + half * 8;
  const int col  = n_base + l15;
#pragma unroll
  for (int r = 0; r < 8; ++r)
    q[(size_t)(row0 + r) * H_ + col] = c[r];
}

// ---------------------------------------------------------------------------
// Kernel B: scores[b,t] = prev[b,t,:] . q[b,:]  fused with the prev_new copy,
// plus append of dh as row T_ of prev_new (chunk-0 blocks).
// Grid: (32 t-chunks of 64 rows, B). Block 256 = 8 waves; 8 rows per wave.
// ---------------------------------------------------------------------------
__global__ void __launch_bounds__(256)
scores_copy(const float* __restrict__ prev, const float* __restrict__ dh,
            const float* __restrict__ q, float* __restrict__ scores,
            float* __restrict__ outPrev) {
  __shared__ __align__(16) float q_s[H_];
  const int b     = blockIdx.y;
  const int chunk = blockIdx.x;
  for (int i = threadIdx.x; i < H_; i += 256) q_s[i] = q[(size_t)b * H_ + i];
  __syncthreads();

  const int wave = threadIdx.x >> 5;
  const int lane = threadIdx.x & 31;
  const float4* prevb = reinterpret_cast<const float4*>(prev + (size_t)b * T_ * H_);
  float4*       outb  = reinterpret_cast<float4*>(outPrev + (size_t)b * (T_ + 1) * H_);

  for (int j = 0; j < 8; ++j) {
    const int t = chunk * 64 + wave * 8 + j;
    const float4* row  = prevb + (size_t)t * (H_ / 4);
    float4*       orow = outb  + (size_t)t * (H_ / 4);
    float acc = 0.0f;
#pragma unroll
    for (int i = 0; i < 8; ++i) {
      const int idx = lane + 32 * i;              // float4 index 0..255
      const float4 v  = row[idx];
      orow[idx] = v;                              // fused copy into prev_new
      const float4 qv = *reinterpret_cast<const float4*>(&q_s[idx * 4]);
      acc += v.x * qv.x + v.y * qv.y + v.z * qv.z + v.w * qv.w;
    }
#pragma unroll
    for (int off = 16; off > 0; off >>= 1)
      acc += __shfl_down(acc, off, 32);
    if (lane == 0) scores[(size_t)b * T_ + t] = acc;
  }

  if (chunk == 0) {  // prev_new[b, T_, :] = dh[b, :]
    float* tail = outPrev + (size_t)b * (T_ + 1) * H_ + (size_t)T_ * H_;
    for (int i = threadIdx.x; i < H_; i += 256) tail[i] = dh[(size_t)b * H_ + i];
  }
}

// ---------------------------------------------------------------------------
// Kernel C: in-place softmax over T_ per batch. One block per batch row.
// ---------------------------------------------------------------------------
__global__ void __launch_bounds__(256)
softmax_rows(float* __restrict__ scores) {
  __shared__ float red[256];
  const int b   = blockIdx.x;
  const int tid = threadIdx.x;
  float* s = scores + (size_t)b * T_;

  float m = -3.4e38f;
  for (int i = tid; i < T_; i += 256) m = fmaxf(m, s[i]);
  red[tid] = m; __syncthreads();
  for (int w = 128; w > 0; w >>= 1) {
    if (tid < w) red[tid] = fmaxf(red[tid], red[tid + w]);
    __syncthreads();
  }
  const float mx = red[0]; __syncthreads();

  float sum = 0.0f;
  for (int i = tid; i < T_; i += 256) {
    const float e = __expf(s[i] - mx);
    s[i] = e;
    sum += e;
  }
  red[tid] = sum; __syncthreads();
  for (int w = 128; w > 0; w >>= 1) {
    if (tid < w) red[tid] += red[tid + w];
    __syncthreads();
  }
  const float inv = 1.0f / red[0];
  for (int i = tid; i < T_; i += 256) s[i] *= inv;
}

// ---------------------------------------------------------------------------
// Kernel D: partial context. Grid (8 t-chunks of 256 rows, B), block 256.
// Thread owns one float4 column; block streams full 4KB rows coalesced.
// Deterministic: partials go to workspace, no float atomics.
// ---------------------------------------------------------------------------
__global__ void __launch_bounds__(256)
context_part(const float* __restrict__ prev, const float* __restrict__ attn,
             float* __restrict__ part) {
  __shared__ float a_s[256];
  const int b     = blockIdx.y;
  const int chunk = blockIdx.x;
  a_s[threadIdx.x] = attn[(size_t)b * T_ + chunk * 256 + threadIdx.x];
  __syncthreads();

  const float4* prevb =
      reinterpret_cast<const float4*>(prev + ((size_t)b * T_ + (size_t)chunk * 256) * H_);
  const int c4 = threadIdx.x;                 // float4 column 0..255
  float ax = 0.f, ay = 0.f, az = 0.f, aw = 0.f;
#pragma unroll 4
  for (int t = 0; t < 256; ++t) {
    const float4 v = prevb[(size_t)t * (H_ / 4) + c4];
    const float  w = a_s[t];
    ax += w * v.x; ay += w * v.y; az += w * v.z; aw += w * v.w;
  }
  float* dst = part + ((size_t)b * 8 + chunk) * H_ + (size_t)c4 * 4;
  dst[0] = ax; dst[1] = ay; dst[2] = az; dst[3] = aw;
}

// ---------------------------------------------------------------------------
// Kernel E: fixed-order reduction of 8 partials -> context output [1,B,H].
// ---------------------------------------------------------------------------
__global__ void __launch_bounds__(256)
context_reduce(const float* __restrict__ part, float* __restrict__ ctx) {
  const int b = blockIdx.x;
  for (int h = threadIdx.x; h < H_; h += 256) {
    float s = 0.0f;
#pragma unroll
    for (int c = 0; c < 8; ++c)
      s += part[((size_t)b * 8 + c) * H_ + h];
    ctx[(size_t)b * H_ + h] = s;
  }
}

extern "C" void kernel_launch(void* const* d_in, const int* in_sizes, int n_in,
                              void* d_out, int out_size, void* d_ws, size_t ws_size,
                              hipStream_t stream) {
  const float* dh   = (const float*)d_in[0];   // [1,B,H] -> [B,H]
  const float* prev = (const float*)d_in[1];   // [B,T,H]
  const float* W    = (const float*)d_in[2];   // [H(out),H(in)] row-major

  float* out     = (float*)d_out;
  float* ctx     = out;                              // [1,B,H] = 32768 floats
  float* outPrev = out + (size_t)B_ * H_;            // [B,T+1,H]

  float* q      = (float*)d_ws;                      // 32*1024
  float* scores = q + (size_t)B_ * H_;               // 32*2048
  float* part   = scores + (size_t)B_ * T_;          // 32*8*1024

  qgemm_wmma<<<128, 32, 0, stream>>>(dh, W, q);
  scores_copy<<<dim3(32, B_), 256, 0, stream>>>(prev, dh, q, scores, outPrev);
  softmax_rows<<<B_, 256, 0, stream>>>(scores);
  context_part<<<dim3(8, B_), 256, 0, stream>>>(prev, scores, part);
  context_reduce<<<B_, 256, 0, stream>>>(part, ctx);
}